// CBandCC_3839700762637
// MI455X (gfx1250) — compile-verified
//
#include <hip/hip_runtime.h>
#include <cstdint>
#include <cstddef>

// ---------------------------------------------------------------------------
// Soft 2D histograms (raised-cosine bilinear splat) + per-histogram max norm.
//   X: (8, 3, 512, 512) f32 in [0,255]  ->  out: (8, 6, 256, 256) f32
//
// MI455X strategy:
//  * One FULL 256x256 f32 histogram (256 KB) privatized in LDS per workgroup —
//    only possible because CDNA5 gives 320 KB LDS per WGP. Accumulate with
//    native no-return ds_add_f32 atomics (uniform bins -> few bank conflicts).
//  * 8 segments per histogram (384 WGs x 1024 thr = 32 wave32 waves) merged
//    via global_atomic_add_f32 (no-return, STOREcnt-tracked).
//  * Normalize kernel: stage the whole 256 KB histogram to LDS with
//    global_load_async_to_lds_b128 (ASYNCcnt), wave32 __shfl_xor max tree,
//    single HBM read + write per element, b128-vectorized throughout.
//  * WMMA considered (H = U^T V rank-N one-hot update) and rejected: 2/256
//    density makes dense tensor-core math ~100x more work than LDS atomics.
// ---------------------------------------------------------------------------

#define NBINS 256
#define NB2   65536       // 256*256
#define NB2V  16384       // NB2 / 4 (float4)
#define BATCH 8
#define NHIST 6
#define IMGHW 262144      // 512*512
#define NCB   261121      // 511*511
#define SEG   8
#define TPB   1024

__global__ void __launch_bounds__(TPB)
zero_out_kernel(float4* __restrict__ out, int n4) {
  int i = blockIdx.x * blockDim.x + threadIdx.x;
  if (i < n4) out[i] = make_float4(0.f, 0.f, 0.f, 0.f);
}

__device__ __forceinline__ void rc_weights(float v, int& i0, float& w0, float& w1) {
  v = fminf(fmaxf(v, 0.0f), 255.0f);
  float f0 = fminf(floorf(v), 254.0f);
  float fr = v - f0;
  w0 = 0.5f * (1.0f + __cosf(3.14159265358979323846f * fr)); // v_cos_f32
  w1 = 1.0f - w0;
  i0 = (int)f0;
}

__device__ __forceinline__ void splat4(float* __restrict__ hist,
                                       float a, float c) {
  int ix, iy;
  float wx0, wx1, wy0, wy1;
  rc_weights(a, ix, wx0, wx1);
  rc_weights(c, iy, wy0, wy1);
  float* hrow = hist + ix * NBINS + iy;
  // LDS float atomics -> ds_add_f32 (no return)
  unsafeAtomicAdd(hrow,             wx0 * wy0);
  unsafeAtomicAdd(hrow + 1,         wx0 * wy1);
  unsafeAtomicAdd(hrow + NBINS,     wx1 * wy0);
  unsafeAtomicAdd(hrow + NBINS + 1, wx1 * wy1);
}

__global__ void __launch_bounds__(TPB)
hist_accum_kernel(const float* __restrict__ X, float* __restrict__ out) {
  extern __shared__ float hist[];           // NB2 floats = 256 KB dynamic LDS
  const int bid = blockIdx.x;
  const int seg = bid % SEG;
  const int ch  = (bid / SEG) % NHIST;
  const int b   = bid / (SEG * NHIST);
  const int tid = threadIdx.x;

  // Vectorized LDS zero-fill (ds_store_b128, 16 iters/thread).
  {
    float4* h4 = (float4*)hist;
    const float4 z = make_float4(0.f, 0.f, 0.f, 0.f);
    for (int i = tid; i < NB2V; i += TPB) h4[i] = z;
  }
  __syncthreads();

  const int ntot = (ch < 3) ? NCB : IMGHW;
  const int per  = (ntot + SEG - 1) / SEG;
  const int lo   = seg * per;
  const int hi   = min(ntot, lo + per);

  if (ch < 3) {
    // cband: pair (X[b,ch,y,x], X[b,ch,y+1,x+1]), idx = y*511 + x
    const float* base = X + (size_t)(b * 3 + ch) * IMGHW;
    if (lo + tid < hi) {
      int y0 = (lo + tid) / 511, x0 = (lo + tid) - y0 * 511;
      __builtin_prefetch(base + y0 * 512 + x0, 0, 1);   // global_prefetch_b8
    }
    for (int idx = lo + tid; idx < hi; idx += TPB) {
      int y = idx / 511;
      int x = idx - y * 511;
      int off = y * 512 + x;
      float a = base[off];
      float c = base[off + 513];                        // (y+1, x+1)
      splat4(hist, a, c);
    }
  } else {
    // cc: RGB_PAIRS = (0,1),(0,2),(1,2) for ch = 3,4,5
    const int ci = (ch == 5) ? 1 : 0;
    const int cj = (ch == 3) ? 1 : 2;
    const float* pa = X + (size_t)(b * 3 + ci) * IMGHW;
    const float* pb = X + (size_t)(b * 3 + cj) * IMGHW;
    if (lo + tid < hi) {
      __builtin_prefetch(pa + lo + tid, 0, 1);
      __builtin_prefetch(pb + lo + tid, 0, 1);
    }
    for (int idx = lo + tid; idx < hi; idx += TPB) {
      splat4(hist, pa[idx], pb[idx]);
    }
  }
  __syncthreads();

  // Merge private LDS histogram into global accumulator.
  // b128 LDS reads; skip-zero test prunes global atomics.
  {
    const float4* h4 = (const float4*)hist;
    float* oh = out + (size_t)(b * NHIST + ch) * NB2;
    for (int i = tid; i < NB2V; i += TPB) {
      float4 v = h4[i];
      float* o = oh + i * 4;
      if (v.x != 0.0f) unsafeAtomicAdd(o,     v.x);     // global_atomic_add_f32
      if (v.y != 0.0f) unsafeAtomicAdd(o + 1, v.y);
      if (v.z != 0.0f) unsafeAtomicAdd(o + 2, v.z);
      if (v.w != 0.0f) unsafeAtomicAdd(o + 3, v.w);
    }
  }
}

__global__ void __launch_bounds__(TPB)
hist_norm_kernel(float* __restrict__ out) {
  // Dynamic LDS: [0, NB2) histogram copy, [NB2, NB2+32) reduction scratch.
  extern __shared__ float lds[];
  const int hid = blockIdx.x;                 // 0..47 = (b*6 + ch)
  const int tid = threadIdx.x;
  float* red = lds + NB2;

  // Stage the whole 256 KB histogram on-chip with async global->LDS b128
  // copies (16 B/lane-transfer, 16 issues/thread). LDS byte offset equals
  // the global byte offset, so one VGPR serves both address fields.
  {
    unsigned long long gbase =
        (unsigned long long)(uintptr_t)(out + (size_t)hid * NB2);
#pragma unroll 4
    for (int k = 0; k < NB2V / TPB; ++k) {
      unsigned off = (unsigned)((tid + k * TPB) * 16);
      asm volatile("global_load_async_to_lds_b128 %0, %1, %2"
                   :: "v"(off), "v"(off), "s"(gbase)
                   : "memory");
    }
    asm volatile("s_wait_asynccnt 0" ::: "memory");
  }
  __syncthreads();

  // Per-thread max over 16 float4s (ds_load_b128), then wave32 shuffle tree,
  // then cross-wave via 32-slot LDS scratch.
  const float4* l4 = (const float4*)lds;
  float m = 0.0f;
  for (int k = tid; k < NB2V; k += TPB) {
    float4 v = l4[k];
    m = fmaxf(m, fmaxf(fmaxf(v.x, v.y), fmaxf(v.z, v.w)));
  }
#pragma unroll
  for (int d = 16; d >= 1; d >>= 1) m = fmaxf(m, __shfl_xor(m, d, 32));
  if ((tid & 31) == 0) red[tid >> 5] = m;
  __syncthreads();
  if (tid < 32) {
    float r = red[tid];
#pragma unroll
    for (int d = 16; d >= 1; d >>= 1) r = fmaxf(r, __shfl_xor(r, d, 32));
    if (tid == 0) red[0] = r;
  }
  __syncthreads();

  const float scale = 1.0f / red[0];
  float4* oh4 = (float4*)(out + (size_t)hid * NB2);
  for (int k = tid; k < NB2V; k += TPB) {
    float4 v = l4[k];
    v.x *= scale; v.y *= scale; v.z *= scale; v.w *= scale;
    oh4[k] = v;                                          // global_store_b128
  }
}

extern "C" void kernel_launch(void* const* d_in, const int* in_sizes, int n_in,
                              void* d_out, int out_size, void* d_ws, size_t ws_size,
                              hipStream_t stream) {
  (void)in_sizes; (void)n_in; (void)d_ws; (void)ws_size;
  const float* X  = (const float*)d_in[0];   // (8,3,512,512) f32
  float* out      = (float*)d_out;           // (8,6,256,256) f32

  // 1) Zero the global accumulators (harness poisons d_out). out_size is an
  //    exact multiple of 4*TPB (3,145,728 = 768 * 4096).
  const int n4 = out_size / 4;
  zero_out_kernel<<<(n4 + TPB - 1) / TPB, TPB, 0, stream>>>((float4*)out, n4);

  // 2) Privatized 256 KB LDS histograms, 8 segments per (b,ch), atomic merge.
  hist_accum_kernel<<<BATCH * NHIST * SEG, TPB, NB2 * sizeof(float), stream>>>(X, out);

  // 3) Async-stage each histogram to LDS (b128), max-reduce, normalize.
  hist_norm_kernel<<<BATCH * NHIST, TPB, (NB2 + 32) * sizeof(float), stream>>>(out);
}